// SpGraphAttentionLayer_71442486001855
// MI455X (gfx1250) — compile-verified
//
#include <hip/hip_runtime.h>
#include <hip/hip_bf16.h>
#include <math.h>

typedef __attribute__((ext_vector_type(2))) float v2f;
typedef __attribute__((ext_vector_type(8))) float v8f;

#define D_IN   128
#define D_OUT  64
#define ALPHA  0.2f

// ---------- orderable-uint encoding of float (for atomicMax over floats) ----------
__device__ __forceinline__ unsigned enc_f32(float f) {
    unsigned u = __float_as_uint(f);
    return (u & 0x80000000u) ? ~u : (u | 0x80000000u);
}
__device__ __forceinline__ float dec_f32(unsigned u) {
    unsigned b = (u & 0x80000000u) ? (u & 0x7FFFFFFFu) : ~u;
    return __uint_as_float(b);
}

// ---------- zero output accumulator, rowsum, and e_max (enc 0 == lowest) ----------
__global__ void init_kernel(float* __restrict__ out, float* __restrict__ rowsum,
                            unsigned* __restrict__ emax, int n) {
    int i = blockIdx.x * blockDim.x + threadIdx.x;
    if (i < n * D_OUT) out[i] = 0.0f;
    if (i < n) { rowsum[i] = 0.0f; emax[i] = 0u; }
}

// ---------- Wh = X @ W via V_WMMA_F32_16X16X4_F32 ----------
// 8 waves/block, each wave computes one 16-row x 64-col tile of Wh.
// W is staged transposed into LDS with pad stride 130 (conflict-free b64 reads).
__global__ __launch_bounds__(256) void gemm_wmma_kernel(
    const float* __restrict__ X, const float* __restrict__ W,
    float* __restrict__ Wh, int n) {
    __shared__ float Wt[D_OUT][D_IN + 2];   // Wt[col][k], stride 130 floats
    int tid = threadIdx.x;
    for (int i = tid; i < D_IN * D_OUT; i += 256) {
        int k = i >> 6;        // row of W
        int c = i & 63;        // col of W
        Wt[c][k] = W[i];
    }
    __syncthreads();

    int wave = tid >> 5;
    int lane = tid & 31;
    int kh   = lane >> 4;      // K-half select (lanes 16-31 hold K+2,K+3)
    int lo   = lane & 15;
    int row_base = (blockIdx.x * 8 + wave) * 16;
    if (row_base >= n) return;                 // wave-uniform: EXEC stays all-1

    int arow = row_base + lo;
    if (arow >= n) arow = n - 1;               // clamp (N is a multiple of 16 anyway)
    const float* xrow = X + (size_t)arow * D_IN;

    v8f acc0 = {}, acc1 = {}, acc2 = {}, acc3 = {};
    for (int k = 0; k < D_IN; k += 4) {
        int kk = k + kh * 2;
        v2f a;
        a.x = xrow[kk];
        a.y = xrow[kk + 1];
        v2f b0 = *(const v2f*)&Wt[ 0 + lo][kk];
        v2f b1 = *(const v2f*)&Wt[16 + lo][kk];
        v2f b2 = *(const v2f*)&Wt[32 + lo][kk];
        v2f b3 = *(const v2f*)&Wt[48 + lo][kk];
        acc0 = __builtin_amdgcn_wmma_f32_16x16x4_f32(false, a, false, b0, (short)0, acc0, false, false);
        acc1 = __builtin_amdgcn_wmma_f32_16x16x4_f32(false, a, false, b1, (short)0, acc1, false, false);
        acc2 = __builtin_amdgcn_wmma_f32_16x16x4_f32(false, a, false, b2, (short)0, acc2, false, false);
        acc3 = __builtin_amdgcn_wmma_f32_16x16x4_f32(false, a, false, b3, (short)0, acc3, false, false);
    }

    // C/D layout: VGPR v, lanes 0-15 -> (M=v, N=lo); lanes 16-31 -> (M=v+8, N=lo)
    int mbase = row_base + kh * 8;
    #pragma unroll
    for (int v = 0; v < 8; ++v) {
        int m = mbase + v;
        if (m < n) {
            float* dst = Wh + (size_t)m * D_OUT + lo;
            dst[ 0] = acc0[v];
            dst[16] = acc1[v];
            dst[32] = acc2[v];
            dst[48] = acc3[v];
        }
    }
}

// ---------- s_src = Wh @ a[:64], s_dst = Wh @ a[64:] ----------
__global__ void score_kernel(const float* __restrict__ Wh, const float* __restrict__ a,
                             float* __restrict__ s_src, float* __restrict__ s_dst, int n) {
    int i = blockIdx.x * blockDim.x + threadIdx.x;
    if (i >= n) return;
    const float* wh = Wh + (size_t)i * D_OUT;
    float ss = 0.0f, sd = 0.0f;
    #pragma unroll
    for (int c = 0; c < D_OUT; ++c) {
        float v = wh[c];
        ss += v * a[c];            // a[] uniform -> scalar loads
        sd += v * a[D_OUT + c];
    }
    s_src[i] = ss;
    s_dst[i] = sd;
}

// ---------- e = leakyrelu(s_src[row] + s_dst[col]); segment max over rows ----------
__global__ void edge_max_kernel(const int* __restrict__ er, const int* __restrict__ ec,
                                const float* __restrict__ s_src, const float* __restrict__ s_dst,
                                float* __restrict__ ebuf, unsigned* __restrict__ emax, int E) {
    int i = blockIdx.x * blockDim.x + threadIdx.x;
    if (i >= E) return;
    int r = er[i], c = ec[i];
    float e = s_src[r] + s_dst[c];
    e = (e > 0.0f) ? e : ALPHA * e;
    ebuf[i] = e;
    atomicMax(emax + r, enc_f32(e));
}

// ---------- w = exp(e - emax[row]); rowsum[row] += w ----------
__global__ void edge_exp_kernel(const int* __restrict__ er, const unsigned* __restrict__ emax,
                                float* __restrict__ ebuf, float* __restrict__ rowsum, int E) {
    int i = blockIdx.x * blockDim.x + threadIdx.x;
    if (i >= E) return;
    int r = er[i];
    float w = __expf(ebuf[i] - dec_f32(emax[r]));
    ebuf[i] = w;
    atomicAdd(rowsum + r, w);
}

// ---------- out[row] += w * Wh[col]  (one wave per edge, 2 cols per lane) ----------
__global__ __launch_bounds__(256) void spmm_kernel(
    const int* __restrict__ er, const int* __restrict__ ec,
    const float* __restrict__ w, const float* __restrict__ Wh,
    float* __restrict__ out, int E) {
    int lane = threadIdx.x & 31;
    int edge = (blockIdx.x * 256 + threadIdx.x) >> 5;
    if (edge >= E) return;
    int r = er[edge], c = ec[edge];
    float wv = w[edge];
    const float* src = Wh + (size_t)c * D_OUT;   // L2-resident gather, coalesced 128B
    float*       dst = out + (size_t)r * D_OUT;
    atomicAdd(dst + lane,      wv * src[lane]);
    atomicAdd(dst + 32 + lane, wv * src[32 + lane]);
}

// ---------- out = elu(out / rowsum) ----------
__global__ void finalize_kernel(float* __restrict__ out, const float* __restrict__ rowsum, int n) {
    int i = blockIdx.x * blockDim.x + threadIdx.x;
    if (i >= n * D_OUT) return;
    float v = out[i] / rowsum[i >> 6];           // D_OUT == 64
    out[i] = (v > 0.0f) ? v : (__expf(v) - 1.0f);
}

extern "C" void kernel_launch(void* const* d_in, const int* in_sizes, int n_in,
                              void* d_out, int out_size, void* d_ws, size_t ws_size,
                              hipStream_t stream) {
    const float* X  = (const float*)d_in[0];
    const int*   er = (const int*)  d_in[1];
    const int*   ec = (const int*)  d_in[2];
    const float* W  = (const float*)d_in[3];
    const float* a  = (const float*)d_in[4];
    int N = in_sizes[0] / D_IN;
    int E = in_sizes[1];
    float* out = (float*)d_out;

    // workspace layout (all 4-byte types; ~40 MB total)
    char* p = (char*)d_ws;
    float*    Wh     = (float*)p;    p += (size_t)N * D_OUT * 4;
    float*    s_src  = (float*)p;    p += (size_t)N * 4;
    float*    s_dst  = (float*)p;    p += (size_t)N * 4;
    unsigned* emax   = (unsigned*)p; p += (size_t)N * 4;
    float*    rowsum = (float*)p;    p += (size_t)N * 4;
    float*    ebuf   = (float*)p;    p += (size_t)E * 4;

    int tpb = 256;
    init_kernel<<<(N * D_OUT + tpb - 1) / tpb, tpb, 0, stream>>>(out, rowsum, emax, N);

    int ntiles = (N + 15) / 16;
    gemm_wmma_kernel<<<(ntiles + 7) / 8, tpb, 0, stream>>>(X, W, Wh, N);

    score_kernel<<<(N + tpb - 1) / tpb, tpb, 0, stream>>>(Wh, a, s_src, s_dst, N);

    edge_max_kernel<<<(E + tpb - 1) / tpb, tpb, 0, stream>>>(er, ec, s_src, s_dst, ebuf, emax, E);
    edge_exp_kernel<<<(E + tpb - 1) / tpb, tpb, 0, stream>>>(er, emax, ebuf, rowsum, E);

    spmm_kernel<<<(E + 7) / 8, tpb, 0, stream>>>(er, ec, ebuf, Wh, out, E);

    finalize_kernel<<<(N * D_OUT + tpb - 1) / tpb, tpb, 0, stream>>>(out, rowsum, N);
}